// ChainCRF_31799937860237
// MI455X (gfx1250) — compile-verified
//
#include <hip/hip_runtime.h>
#include <hip/hip_bf16.h>

// Problem sizes (match reference).
#define BB 256
#define TT 2048
#define CC 64

#define L2E 1.4426950408889634f  // log2(e)
#define LN2 0.6931471805599453   // ln(2)

typedef float v2f __attribute__((ext_vector_type(2)));
typedef float v8f __attribute__((ext_vector_type(8)));

// ds_swizzle group-of-32 encoding: offset[14:10]=xor, [9:5]=or, [4:0]=and.
#define SWZ_XOR(x) (((x) << 10) | 0x1f)

template <int PAT>
__device__ __forceinline__ float swz(float v) {
  return __int_as_float(__builtin_amdgcn_ds_swizzle(__float_as_int(v), PAT));
}

// Depth-2 butterfly: after this, lane 4q of each quad holds the quad max.
__device__ __forceinline__ float quadmax4(float v) {
  v = fmaxf(v, swz<SWZ_XOR(1)>(v));
  v = fmaxf(v, swz<SWZ_XOR(2)>(v));
  return v;
}
// Full 16-lane-half butterfly sum (epilogue only).
__device__ __forceinline__ float rowsum16(float v) {
  v += swz<SWZ_XOR(1)>(v);
  v += swz<SWZ_XOR(2)>(v);
  v += swz<SWZ_XOR(4)>(v);
  v += swz<SWZ_XOR(8)>(v);
  return v;
}

// ---------------------------------------------------------------------------
// Kernel 1: path energy per batch:
//   emit  = sum_t x[b,t,y[b,t]] + b_start[y[b,0]] + b_end[y[b,T-1]]
//   trans = sum_{t<T-1} U[y[b,t], y[b,t+1]]
// ---------------------------------------------------------------------------
__global__ __launch_bounds__(256) void path_energy_kernel(
    const float* __restrict__ x, const int* __restrict__ y,
    const float* __restrict__ U, const float* __restrict__ b_start,
    const float* __restrict__ b_end, float* __restrict__ path) {
  const int b = blockIdx.x;
  const int tid = threadIdx.x;
  const int* yb = y + (size_t)b * TT;
  const float* xb = x + (size_t)b * TT * CC;

  float acc = 0.f;
  for (int t = tid; t < TT; t += 256) {
    const int yt = yb[t];
    acc += xb[(size_t)t * CC + yt];
    if (t == 0) acc += b_start[yt];
    if (t == TT - 1) acc += b_end[yt];
    if (t < TT - 1) acc += U[yt * CC + yb[t + 1]];
  }

  __shared__ float red[256];
  red[tid] = acc;
  __syncthreads();
  for (int s = 128; s > 0; s >>= 1) {
    if (tid < s) red[tid] += red[tid + s];
    __syncthreads();
  }
  if (tid == 0) path[b] = red[0];
}

// ---------------------------------------------------------------------------
// Kernel 2: scaled CRF forward scan via f32 WMMA, exp-domain state.
// Grid: 16 blocks (16 batches). Block: 128 threads = 4 waves; wave w owns
// output states [16w,16w+16). State e = 2^(alpha*log2e - carry) lives in LDS;
// per step each wave does 16x16 <- 16x64 @ 64x16 as 16 chained
// V_WMMA_F32_16X16X4_F32 fed DIRECTLY from ds_load_b64 (no VALU between).
// Rescale e' = D * 2^(x*log2e - mu), mu = log2(rowmax(e)), carry += mu.
// ---------------------------------------------------------------------------
__global__ __launch_bounds__(128) void crf_forward_kernel(
    const float* __restrict__ x, const float* __restrict__ U,
    const float* __restrict__ b_start, const float* __restrict__ b_end,
    const float* __restrict__ path, float* __restrict__ out) {
  constexpr int SP = 68;  // padded row stride (floats): 8B-aligned, 0-conflict
  __shared__ __align__(16) float sbuf[2][16 * SP];  // e values, 16 x 64
  __shared__ float pbuf[2][256];                    // quad maxes [w][q][row]
  __shared__ float psum[64];

  const int tid = threadIdx.x;
  const int lane = tid & 31;
  const int w = tid >> 5;            // wave id = N-tile
  const int half = (lane >> 4) & 1;  // 0: lanes 0-15, 1: lanes 16-31
  const int lm = lane & 15;
  const int b0 = blockIdx.x * 16;
  const int col = w * 16 + lm;  // D layout: this lane's output state

  // Loop-invariant B fragments: expU (natural exp), 64(K) x 16(N) as 16
  // chunks of K=4. Assumed 4x16 f32 B layout (dual of documented A 16x4).
  v2f Bf[16];
#pragma unroll
  for (int c = 0; c < 16; ++c) {
    const int k0 = 4 * c + (half ? 2 : 0);
    Bf[c].x = __builtin_amdgcn_exp2f(U[k0 * CC + col] * L2E);
    Bf[c].y = __builtin_amdgcn_exp2f(U[(k0 + 1) * CC + col] * L2E);
  }

  const float bs2 = b_start[col] * L2E;
  const float be2 = b_end[col] * L2E;

  // D-layout x tile pointer: reg r -> batch row (r + half*8), column `col`.
  const float* xb = x + (size_t)(b0 + half * 8) * (TT * CC) + col;

  // e_0 = 2^((x0 + b_start) * log2e), in C/D fragment layout.
  float ereg[8];
#pragma unroll
  for (int r = 0; r < 8; ++r)
    ereg[r] = __builtin_amdgcn_exp2f(fmaf(xb[(size_t)r * TT * CC], L2E, bs2));
  xb += CC;  // now at t=1

  // Publish e_0 + depth-2 row-max partials.
#pragma unroll
  for (int r = 0; r < 8; ++r) sbuf[0][(r + half * 8) * SP + col] = ereg[r];
  {
    float pm[8];
#pragma unroll
    for (int r = 0; r < 8; ++r) pm[r] = quadmax4(ereg[r]);
    if ((lm & 3) == 0) {
      const int q = lm >> 2;
#pragma unroll
      for (int r = 0; r < 8; ++r)
        pbuf[0][w * 64 + q * 16 + (r + half * 8)] = pm[r];
    }
  }
  __syncthreads();

  // A-read base: lane reads batch row m = lane%16, K pair (4c + half*2).
  const int aoff = lm * SP + (half ? 2 : 0);

  // Depth-1 software pipeline for x.
  float xv[8];
#pragma unroll
  for (int r = 0; r < 8; ++r) xv[r] = xb[(size_t)r * TT * CC];
  xb += CC;

  double c_acc = 0.0;  // base-2 carry for row (lane%16), f64 for accuracy
  int cur = 0;

#pragma unroll 1
  for (int t = 1; t < TT; ++t) {
    // Prefetch ~15 steps ahead into L2 (speculative; OOB tail is dropped).
#pragma unroll
    for (int r = 0; r < 8; ++r)
      __builtin_prefetch(&xb[(size_t)r * TT * CC + 14 * CC], 0, 1);
    // Load x for step t+1 (consumed next iteration).
    float xn[8];
    if (t < TT - 1) {
#pragma unroll
      for (int r = 0; r < 8; ++r) xn[r] = xb[(size_t)r * TT * CC];
      xb += CC;
    }

    // D = e @ expU: pure LDS -> WMMA chain over K=64 in steps of 4.
    v8f acc = {0.f, 0.f, 0.f, 0.f, 0.f, 0.f, 0.f, 0.f};
#pragma unroll
    for (int c = 0; c < 16; ++c) {
      const v2f a = *(const v2f*)&sbuf[cur][aoff + 4 * c];
      acc = __builtin_amdgcn_wmma_f32_16x16x4_f32(
          false, a, false, Bf[c], (short)0, acc, false, false);
    }

    // Row rescale mu = log2(max_i e[row,i]): 16-way combine of quad maxes
    // (overlaps the WMMA chain; only needed at write time).
    float mu_e = pbuf[cur][lm];
#pragma unroll
    for (int i = 1; i < 16; ++i) mu_e = fmaxf(mu_e, pbuf[cur][i * 16 + lm]);
    const float mu = __builtin_amdgcn_logf(mu_e);  // v_log_f32 == log2
    c_acc += (double)mu;

    // e' = D * 2^(x*log2e + (b_end?)*log2e - mu).
    const float shift = ((t == TT - 1) ? be2 : 0.f) - mu;
#pragma unroll
    for (int r = 0; r < 8; ++r)
      ereg[r] = acc[r] * __builtin_amdgcn_exp2f(fmaf(xv[r], L2E, shift));

    // Publish e' + depth-2 quad maxes into the other buffer; one barrier.
    const int nxt = cur ^ 1;
#pragma unroll
    for (int r = 0; r < 8; ++r) sbuf[nxt][(r + half * 8) * SP + col] = ereg[r];
    {
      float pm[8];
#pragma unroll
      for (int r = 0; r < 8; ++r) pm[r] = quadmax4(ereg[r]);
      if ((lm & 3) == 0) {
        const int q = lm >> 2;
#pragma unroll
        for (int r = 0; r < 8; ++r)
          pbuf[nxt][w * 64 + q * 16 + (r + half * 8)] = pm[r];
      }
    }
    __syncthreads();
    cur = nxt;

#pragma unroll
    for (int r = 0; r < 8; ++r) xv[r] = xn[r];
  }

  // Epilogue: log_Z[b] = ln2 * (carry_b + log2(sum_j e_T[b,j])).
  float ps[8];
#pragma unroll
  for (int r = 0; r < 8; ++r) ps[r] = rowsum16(ereg[r]);
  if (lm == 0) {
#pragma unroll
    for (int r = 0; r < 8; ++r) psum[w * 16 + r + half * 8] = ps[r];
  }
  __syncthreads();

  if (tid < 16) {
    const float S = psum[tid] + psum[16 + tid] + psum[32 + tid] + psum[48 + tid];
    const double lz = (c_acc + (double)__builtin_amdgcn_logf(S)) * LN2;
    out[b0 + tid] = (float)lz - path[b0 + tid];
  }
}

// ---------------------------------------------------------------------------
extern "C" void kernel_launch(void* const* d_in, const int* in_sizes, int n_in,
                              void* d_out, int out_size, void* d_ws,
                              size_t ws_size, hipStream_t stream) {
  const float* x = (const float*)d_in[0];
  const int* y = (const int*)d_in[1];
  const float* U = (const float*)d_in[2];
  const float* bs = (const float*)d_in[3];
  const float* be = (const float*)d_in[4];
  float* path = (float*)d_ws;  // 256 floats of scratch
  float* out = (float*)d_out;  // 256 floats (B,1)

  path_energy_kernel<<<BB, 256, 0, stream>>>(x, y, U, bs, be, path);
  crf_forward_kernel<<<BB / 16, 128, 0, stream>>>(x, U, bs, be, path, out);

  (void)in_sizes; (void)n_in; (void)out_size; (void)ws_size;
}